// RFGridSampleDAS_53223234732543
// MI455X (gfx1250) — compile-verified
//
#include <hip/hip_runtime.h>
#include <stdint.h>

// ---- problem geometry (fixed by the reference) -----------------------------
namespace {
constexpr int kA = 4;              // angles
constexpr int kE = 128;            // elements
constexpr int kS = 3072;           // samples per trace
constexpr int kNZ = 512, kNX = 256;
constexpr int kNP = kNZ * kNX;     // 131072 pixels
constexpr int kThreads = 256;      // 8 waves (wave32)
constexpr int kPixPerThread = 2;
constexpr int kPixPerBlock = kThreads * kPixPerThread;   // 512
constexpr int kBlocks = kNP / kPixPerBlock;              // 256
constexpr int kBufFloats = kA * kS;                      // 12288 floats = 48 KB
constexpr size_t kLdsBytes = 2ull * kBufFloats * sizeof(float);  // 96 KB double-buffered
}

typedef uint32_t v4u __attribute__((ext_vector_type(4)));
typedef int32_t  v8i __attribute__((ext_vector_type(8)));
typedef int32_t  v4i __attribute__((ext_vector_type(4)));

// Stage rf[0..3, e, 0..S-1] (4 rows, row stride E*S floats) into LDS at lds_off
// using the Tensor Data Mover. D# layout per CDNA5 ISA ch.8.
__device__ __forceinline__ void tdm_load_rf_rows(const float* rf_row0, uint32_t lds_off) {
  const uint64_t ga = (uint64_t)(uintptr_t)rf_row0;
  v4u g0;
  g0[0] = 1u;                                          // count=1, user descriptor
  g0[1] = lds_off;                                     // lds_addr (bytes)
  g0[2] = (uint32_t)(ga & 0xFFFFFFFFu);                // global_addr[31:0]
  g0[3] = (uint32_t)((ga >> 32) & 0x01FFFFFFu)         // global_addr[56:32]
          | (2u << 30);                                // type = 2 ("image")
  v8i g1;
  g1[0] = (int)(2u << 16);                             // data_size = 2 -> 4 bytes
  g1[1] = (int)((uint32_t)kS << 16);                   // tensor_dim0[15:0]=3072
  g1[2] = (int)(((uint32_t)(kA * kE)) << 16);          // tensor_dim0 hi=0 | tensor_dim1 lo
  g1[3] = (int)(((uint32_t)kS) << 16);                 // tensor_dim1 hi=0 | tile_dim0=3072
  g1[4] = (int)kA;                                     // tile_dim1=4, tile_dim2=0
  g1[5] = (int)(kE * kS);                              // tensor_dim0_stride lo32 = 393216
  g1[6] = 0;                                           // stride0 hi | stride1 lo
  g1[7] = 0;                                           // stride1 hi
  const v4i z4 = {0, 0, 0, 0};                         // groups 2/3 unused (2D tile)
  const v8i z8 = {0, 0, 0, 0, 0, 0, 0, 0};             // extra group (clang-23 6-arg form)
  __builtin_amdgcn_tensor_load_to_lds(g0, g1, z4, z4, z8, 0);
}

__global__ __launch_bounds__(kThreads)
void RFGridSampleDAS_53223234732543_kernel(
    const float* __restrict__ d_tx,   // [A, NZ*NX]
    const float* __restrict__ d_rx,   // [E, NZ*NX]
    const float* __restrict__ apod,   // [E, NZ*NX]
    const float* __restrict__ rf,     // [A, E, S]
    const float* __restrict__ t0,     // [A]
    float* __restrict__ out) {        // [A, NZ*NX]
  extern __shared__ float smem[];     // 2 x (A * S) floats, double-buffered
  const int tid = threadIdx.x;
  const int p0  = blockIdx.x * kPixPerBlock + tid;
  const int p1  = p0 + kThreads;

  // Low 32 bits of a flat LDS pointer are the wave-relative LDS byte offset.
  const uint32_t lds_base = (uint32_t)(uint64_t)(uintptr_t)smem;

  // Per-pixel tx-delay (already minus t0) kept in registers for all 4 angles.
  float dtx0[kA], dtx1[kA];
#pragma unroll
  for (int a = 0; a < kA; ++a) {
    const float t = t0[a];
    dtx0[a] = d_tx[a * kNP + p0] - t;
    dtx1[a] = d_tx[a * kNP + p1] - t;
  }
  float acc0[kA] = {0.f, 0.f, 0.f, 0.f};
  float acc1[kA] = {0.f, 0.f, 0.f, 0.f};

  // Wave-uniform scalar guard so exactly one wave drives the TDM.
  const int wave_id = __builtin_amdgcn_readfirstlane(tid) >> 5;
  const bool issuer = (wave_id == 0);

  if (issuer) {
    tdm_load_rf_rows(rf /* + 0*kS */, lds_base);       // prefetch e = 0 -> buf 0
  }

  for (int e = 0; e < kE; ++e) {
    const int cur = e & 1;
    if (issuer) {
      if (e + 1 < kE) {
        tdm_load_rf_rows(rf + (size_t)(e + 1) * kS,
                         lds_base + (uint32_t)((cur ^ 1) * kBufFloats * 4));
        __builtin_amdgcn_s_wait_tensorcnt(1);          // tile e complete (in-order)
      } else {
        __builtin_amdgcn_s_wait_tensorcnt(0);
      }
    }
    __syncthreads();                                   // tile e visible to all waves

    const float* __restrict__ buf = smem + cur * kBufFloats;
    // Streamed exactly once device-wide: non-temporal so rf stays L2-resident.
    const float drx0 = __builtin_nontemporal_load(d_rx + e * kNP + p0);
    const float drx1 = __builtin_nontemporal_load(d_rx + e * kNP + p1);
    const float ap0  = __builtin_nontemporal_load(apod + e * kNP + p0);
    const float ap1  = __builtin_nontemporal_load(apod + e * kNP + p1);

#pragma unroll
    for (int a = 0; a < kA; ++a) {
      const float* __restrict__ row = buf + a * kS;
      {
        const float dl = dtx0[a] + drx0;
        const float x0 = floorf(dl);
        const float w  = dl - x0;
        const int   i0 = (int)x0;
        const float v0 = (i0 >= 0 && i0 < kS)         ? row[i0]     : 0.f;
        const float v1 = (i0 >= -1 && i0 < kS - 1)    ? row[i0 + 1] : 0.f;
        acc0[a] += (v0 + (v1 - v0) * w) * ap0;
      }
      {
        const float dl = dtx1[a] + drx1;
        const float x0 = floorf(dl);
        const float w  = dl - x0;
        const int   i0 = (int)x0;
        const float v0 = (i0 >= 0 && i0 < kS)         ? row[i0]     : 0.f;
        const float v1 = (i0 >= -1 && i0 < kS - 1)    ? row[i0 + 1] : 0.f;
        acc1[a] += (v0 + (v1 - v0) * w) * ap1;
      }
    }
    __syncthreads();                                   // done reading buf[cur] before reuse
  }

#pragma unroll
  for (int a = 0; a < kA; ++a) {
    __builtin_nontemporal_store(acc0[a], out + a * kNP + p0);
    __builtin_nontemporal_store(acc1[a], out + a * kNP + p1);
  }
}

extern "C" void kernel_launch(void* const* d_in, const int* in_sizes, int n_in,
                              void* d_out, int out_size, void* d_ws, size_t ws_size,
                              hipStream_t stream) {
  (void)in_sizes; (void)n_in; (void)out_size; (void)d_ws; (void)ws_size;
  const float* d_tx = (const float*)d_in[0];
  const float* d_rx = (const float*)d_in[1];
  const float* apod = (const float*)d_in[2];
  const float* rf   = (const float*)d_in[3];
  const float* t0   = (const float*)d_in[4];
  float* out = (float*)d_out;

  dim3 grid(kBlocks), block(kThreads);
  RFGridSampleDAS_53223234732543_kernel<<<grid, block, kLdsBytes, stream>>>(
      d_tx, d_rx, apod, rf, t0, out);
}